// CustomGATLayerIsotropic_25632364822810
// MI455X (gfx1250) — compile-verified
//
#include <hip/hip_runtime.h>
#include <hip/hip_bf16.h>
#include <math.h>

// ---------------------------------------------------------------------------
// GAT-isotropic layer for MI455X (gfx1250, wave32):
//   z   = h @ Wf^T                (WMMA f32 16x16x4, fp32 precision)
//   agg = segment_sum(z[src], dst)  (atomic scatter-add, HBM-bound)
//   out = h + elu(BN(agg))          (streaming fused epilogue)
// ---------------------------------------------------------------------------

typedef float v2f __attribute__((ext_vector_type(2)));
typedef float v8f __attribute__((ext_vector_type(8)));

#define IN_DIM 256
#define HXO    256   // H * OUT_DIM
#define LDS_STRIDE 260  // 256 + 4 pad -> bank(k + 4*row), conflict-free

// ---------------------------------------------------------------------------
// Kernel 1: z[n, m] = sum_d h[n,d] * Wf[m,d],  Wf = W reshaped [H*O, D]
// One block (8 waves) per 16-row tile. Each wave owns two 16-col tiles.
// ---------------------------------------------------------------------------
__global__ __launch_bounds__(256) void gemm_wmma_f32(
    const float* __restrict__ h, const float* __restrict__ Wf,
    float* __restrict__ z) {
  __shared__ float lds[16 * LDS_STRIDE];

  const int tid   = threadIdx.x;
  const int rbase = blockIdx.x * 16;

  // Stage 16x256 tile of h into LDS (coalesced global, conflict-free DS).
  for (int i = tid; i < 16 * IN_DIM; i += 256) {
    const int r = i >> 8;
    const int c = i & 255;
    lds[r * LDS_STRIDE + c] = h[(rbase + r) * IN_DIM + c];
  }
  __syncthreads();

  const int wave = tid >> 5;
  const int lane = tid & 31;
  const int row  = lane & 15;            // M for A-frag, N for B-frag
  const int kgrp = (lane >> 4) << 1;     // lanes 0-15 -> K 0..1, 16-31 -> K 2..3

  const int c0    = wave * 32;           // this wave's first output column
  const float* lrow = &lds[row * LDS_STRIDE + kgrp];
  const float* w0   = Wf + (c0 + row)      * IN_DIM + kgrp;
  const float* w1   = Wf + (c0 + 16 + row) * IN_DIM + kgrp;

  v8f acc0 = {};
  v8f acc1 = {};

#pragma unroll 8
  for (int k0 = 0; k0 < IN_DIM; k0 += 4) {
    v2f a;
    a.x = lrow[k0];
    a.y = lrow[k0 + 1];
    v2f b0 = *(const v2f*)(w0 + k0);
    v2f b1 = *(const v2f*)(w1 + k0);
    // 8-arg pattern: (neg_a, A, neg_b, B, c_mod, C, reuse_a, reuse_b)
    acc0 = __builtin_amdgcn_wmma_f32_16x16x4_f32(false, a, false, b0,
                                                 (short)0, acc0, false, false);
    acc1 = __builtin_amdgcn_wmma_f32_16x16x4_f32(false, a, false, b1,
                                                 (short)0, acc1, false, false);
  }

  // C/D layout: VGPR v, lanes 0-15 -> M=v, lanes 16-31 -> M=v+8, N=lane&15.
  const int rtop = rbase + ((lane >> 4) << 3);
  const int colb = c0 + row;
#pragma unroll
  for (int v = 0; v < 8; ++v) {
    z[(rtop + v) * HXO + colb]      = acc0[v];
    z[(rtop + v) * HXO + colb + 16] = acc1[v];
  }
}

// ---------------------------------------------------------------------------
// Kernel 2: zero agg buffer + stats accumulators (deterministic per call)
// ---------------------------------------------------------------------------
__global__ __launch_bounds__(256) void init_zero(float* __restrict__ p,
                                                 long n) {
  long i = (long)blockIdx.x * blockDim.x + threadIdx.x;
  long stride = (long)gridDim.x * blockDim.x;
  for (; i < n; i += stride) p[i] = 0.0f;
}

// ---------------------------------------------------------------------------
// Kernel 3: agg[dst[e], :] += z[src[e], :]   (thread = channel, block = edge)
// ---------------------------------------------------------------------------
__global__ __launch_bounds__(256) void scatter_add_edges(
    const float* __restrict__ z, const int* __restrict__ src,
    const int* __restrict__ dst, float* __restrict__ agg, int E) {
  const int c = threadIdx.x;
  for (int e = blockIdx.x; e < E; e += gridDim.x) {
    const int s = src[e];
    const int d = dst[e];
    const float v = z[(long)s * HXO + c];
    atomicAdd(&agg[(long)d * HXO + c], v);
  }
}

// ---------------------------------------------------------------------------
// Kernel 4: per-channel sum / sumsq over N rows (thread = channel)
// ---------------------------------------------------------------------------
__global__ __launch_bounds__(256) void bn_stats(
    const float* __restrict__ agg, float* __restrict__ sums,
    float* __restrict__ sumsq, int N) {
  const int c = threadIdx.x;
  float s = 0.0f, s2 = 0.0f;
  for (int n = blockIdx.x; n < N; n += gridDim.x) {
    const float v = agg[(long)n * HXO + c];
    s += v;
    s2 += v * v;
  }
  atomicAdd(&sums[c], s);
  atomicAdd(&sumsq[c], s2);
}

// ---------------------------------------------------------------------------
// Kernel 5: fold BN into per-channel scale/bias (1 block of 256)
// ---------------------------------------------------------------------------
__global__ __launch_bounds__(256) void bn_params(
    const float* __restrict__ sums, const float* __restrict__ sumsq,
    const float* __restrict__ gamma, const float* __restrict__ beta,
    float* __restrict__ scale, float* __restrict__ bias, int N) {
  const int c = threadIdx.x;
  const float inv = 1.0f / (float)N;
  const float mean = sums[c] * inv;
  float var = sumsq[c] * inv - mean * mean;   // biased, matches jnp.var
  var = fmaxf(var, 0.0f);
  const float sc = gamma[c] * rsqrtf(var + 1e-5f);
  scale[c] = sc;
  bias[c]  = beta[c] - mean * sc;
}

// ---------------------------------------------------------------------------
// Kernel 6: out = h + elu(agg*scale + bias)   (fully coalesced stream)
// ---------------------------------------------------------------------------
__global__ __launch_bounds__(256) void elu_residual(
    const float* __restrict__ h, const float* __restrict__ agg,
    const float* __restrict__ scale, const float* __restrict__ bias,
    float* __restrict__ out, long total) {
  long i = (long)blockIdx.x * blockDim.x + threadIdx.x;
  long stride = (long)gridDim.x * blockDim.x;
  for (; i < total; i += stride) {
    const int c = (int)(i & 255);
    const float x = agg[i] * scale[c] + bias[c];
    const float y = (x > 0.0f) ? x : expm1f(x);
    out[i] = h[i] + y;
  }
}

// ---------------------------------------------------------------------------
extern "C" void kernel_launch(void* const* d_in, const int* in_sizes, int n_in,
                              void* d_out, int out_size, void* d_ws,
                              size_t ws_size, hipStream_t stream) {
  const float* h     = (const float*)d_in[0];   // [N, 256]
  const float* Wf    = (const float*)d_in[1];   // [H,O,D] -> [256, 256]
  const float* gamma = (const float*)d_in[2];   // [H,O] -> [256]
  const float* beta  = (const float*)d_in[3];   // [H,O] -> [256]
  const int*   src   = (const int*)d_in[4];     // [E]
  const int*   dst   = (const int*)d_in[5];     // [E]
  float*       out   = (float*)d_out;           // [N, 256]

  const int N = in_sizes[0] / IN_DIM;
  const int E = in_sizes[4];

  // Workspace layout
  char* ws = (char*)d_ws;
  const size_t zBytes = (size_t)N * HXO * sizeof(float);
  float* z      = (float*)ws;
  float* agg    = (float*)(ws + zBytes);
  float* sums   = (float*)(ws + 2 * zBytes);
  float* sumsq  = sums + HXO;
  float* scale  = sums + 2 * HXO;
  float* bias   = sums + 3 * HXO;

  const long aggTotal = (long)N * HXO;

  // 1) projection GEMM via WMMA (N assumed multiple of 16; 100000/16 = 6250)
  gemm_wmma_f32<<<N / 16, 256, 0, stream>>>(h, Wf, z);

  // 2) zero agg + stats (agg and 4*256 floats of stats/params are contiguous)
  init_zero<<<2048, 256, 0, stream>>>(agg, aggTotal + 4 * HXO);

  // 3) edge gather + scatter-add
  int sgrid = E < 65536 ? E : 65536;
  scatter_add_edges<<<sgrid, 256, 0, stream>>>(z, src, dst, agg, E);

  // 4) BN batch statistics
  bn_stats<<<512, 256, 0, stream>>>(agg, sums, sumsq, N);

  // 5) fold into scale/bias
  bn_params<<<1, 256, 0, stream>>>(sums, sumsq, gamma, beta, scale, bias, N);

  // 6) normalize + ELU + residual
  elu_residual<<<4096, 256, 0, stream>>>(h, agg, scale, bias, out, aggTotal);
}